// auditorynerve_14104672600179
// MI455X (gfx1250) — compile-verified
//
#include <hip/hip_runtime.h>
#include <stdint.h>

// LIF scan: mem_t = 0.95*mem_{t-1} + x_t - (mem_{t-1} > 1), spk_t = (mem_t > 1)
// N=8192 neurons (independent, parallel), T=4000 steps (sequential).
// Memory-bound (262MB @ 23.3TB/s ~= 11us floor): stream tiles through the
// CDNA5 async global<->LDS DMA path (ASYNCcnt), double-buffered both ways.

#define BETA 0.95f
#define UTH  1.0f

constexpr int BLOCK  = 256;  // 256 neurons per workgroup, 1 thread each (8 waves)
constexpr int TT     = 8;    // time steps per tile (32B per neuron-row per tile)
constexpr int STRIDE = 12;   // padded LDS row floats (48B: 16B-aligned, kills bank conflicts)
constexpr int XFERS  = 2;    // B128 transfers per thread per tile

#if defined(__HIP_DEVICE_COMPILE__)

// Builtin prototypes (probe-confirmed): b128 async DMA builtins take pointers
// to 16-byte int vectors in address spaces 1 (global) and 3 (LDS).
typedef int v4i __attribute__((vector_size(16)));
typedef __attribute__((address_space(1))) v4i* gv4i_t;  // global
typedef __attribute__((address_space(3))) v4i* lv4i_t;  // LDS

#if __has_builtin(__builtin_amdgcn_s_wait_asynccnt)
#define WAIT_ASYNCCNT(n) __builtin_amdgcn_s_wait_asynccnt(n)
#else
#define WAIT_ASYNCCNT(n) asm volatile("s_wait_asynccnt %0" ::"i"(n) : "memory")
#endif

// cpol=1 -> TH=NT (non-temporal): 262MB streamed once, no reuse; keep L2 clean.
__device__ __forceinline__ void async_load_b128_nt(uint64_t gaddr, void* lptr) {
#if __has_builtin(__builtin_amdgcn_global_load_async_to_lds_b128)
  __builtin_amdgcn_global_load_async_to_lds_b128(
      (gv4i_t)(uintptr_t)gaddr, (lv4i_t)lptr, /*offset=*/0, /*cpol=*/1);
#else
  uint32_t l = (uint32_t)(uintptr_t)(lv4i_t)lptr;
  asm volatile("global_load_async_to_lds_b128 %0, %1, off th:TH_LOAD_NT" ::"v"(l), "v"(gaddr)
               : "memory");
#endif
}

__device__ __forceinline__ void async_store_b128_nt(uint64_t gaddr, const void* lptr) {
#if __has_builtin(__builtin_amdgcn_global_store_async_from_lds_b128)
  __builtin_amdgcn_global_store_async_from_lds_b128(
      (gv4i_t)(uintptr_t)gaddr, (lv4i_t)(void*)(uintptr_t)lptr, /*offset=*/0, /*cpol=*/1);
#else
  uint32_t l = (uint32_t)(uintptr_t)(lv4i_t)(void*)(uintptr_t)lptr;
  asm volatile("global_store_async_from_lds_b128 %0, %1, off th:TH_STORE_NT" ::"v"(gaddr), "v"(l)
               : "memory");
#endif
}

#else  // host pass: bodies never emitted
#define WAIT_ASYNCCNT(n) ((void)0)
__device__ __forceinline__ void async_load_b128_nt(uint64_t, void*) {}
__device__ __forceinline__ void async_store_b128_nt(uint64_t, const void*) {}
#endif

// Requires N % BLOCK == 0 and T % (2*TT) == 0 (true for the fixed 8192x4000 shape).
__global__ __launch_bounds__(BLOCK) void lif_scan_kernel(
    const float* __restrict__ inp, float* __restrict__ out, int T) {
  __shared__ float sin_[2][BLOCK * STRIDE];   // input tiles (double-buffered)
  __shared__ float sout_[2][BLOCK * STRIDE];  // spike tiles (double-buffered)

  const int tid    = threadIdx.x;
  const int nbase  = blockIdx.x * BLOCK;
  const int ntiles = T / TT;  // 500 (even)

  // Cooperative tile DMA mapping: tile = [256 rows x 8 floats], 2 chunks of 16B
  // per row. Transfer k of this thread handles row = (tid>>1) + k*128, chunk =
  // tid&1, so lane pairs cover contiguous 32B global rows. All bases hoisted;
  // the global side just advances 32B per tile.
  const int row = tid >> 1, chunk = tid & 1;

  const uint64_t gin0 = (uint64_t)(uintptr_t)(inp + (size_t)(nbase + row) * T + chunk * 4);
  const uint64_t gin1 = gin0 + (uint64_t)128 * T * sizeof(float);
  const uint64_t gout0 = (uint64_t)(uintptr_t)(out + (size_t)(nbase + row) * T + chunk * 4);
  const uint64_t gout1 = gout0 + (uint64_t)128 * T * sizeof(float);

  float* const lin0  = &sin_[0][row * STRIDE + chunk * 4];   // +128 rows = +128*STRIDE floats
  float* const lout0 = &sout_[0][row * STRIDE + chunk * 4];
  constexpr int ROWJMP = 128 * STRIDE;          // second transfer's LDS row offset
  constexpr int BUFJMP = BLOCK * STRIDE;        // buffer 1 offset in floats

  auto issue_loads = [&](int it, int b) {
    const uint64_t g = gin0 + (uint64_t)it * (TT * sizeof(float));
    float* l = lin0 + b * BUFJMP;
    async_load_b128_nt(g, l);
    async_load_b128_nt(g + (uint64_t)128 * T * sizeof(float) - (gin0 - gin0), l + ROWJMP);
  };
  auto issue_stores = [&](int it, int b) {
    const uint64_t g = gout0 + (uint64_t)it * (TT * sizeof(float));
    float* l = lout0 + b * BUFJMP;
    async_store_b128_nt(g, l);
    async_store_b128_nt(g + (uint64_t)128 * T * sizeof(float), l + ROWJMP);
  };

  const float* rin0 = &sin_[0][tid * STRIDE];
  float* rout0 = &sout_[0][tid * STRIDE];

  float mem = 0.0f;

  // One pipeline stage: compute tile `it` out of buffer `b` (compile-time 0/1).
  auto step = [&](int it, int b) {
    if (it + 1 < ntiles) issue_loads(it + 1, b ^ 1);

    // Drains all but the 2 loads just issued: tile `it` loads are complete and
    // stores from iteration it-2 (which read sout_[b]) are complete (loads
    // in-order w/ loads, stores in-order w/ stores).
    WAIT_ASYNCCNT(2);
    __syncthreads();  // make all waves' DMA'd LDS tiles visible block-wide

    const float* rin = rin0 + b * BUFJMP;
    const float4 x0 = *(const float4*)(rin);
    const float4 x1 = *(const float4*)(rin + 4);
    const float xs[TT] = {x0.x, x0.y, x0.z, x0.w, x1.x, x1.y, x1.z, x1.w};
    float sp[TT];
#pragma unroll
    for (int t = 0; t < TT; ++t) {
      const float reset = (mem > UTH) ? UTH : 0.0f;  // uses PREVIOUS mem
      mem = __builtin_fmaf(BETA, mem, xs[t]) - reset;
      sp[t] = (mem > UTH) ? 1.0f : 0.0f;
    }
    float* rout = rout0 + b * BUFJMP;
    *(float4*)(rout)     = make_float4(sp[0], sp[1], sp[2], sp[3]);
    *(float4*)(rout + 4) = make_float4(sp[4], sp[5], sp[6], sp[7]);

    __syncthreads();  // spike tile complete before cooperative async store
    issue_stores(it, b);
  };

  issue_loads(0, 0);
  for (int i = 0; i < ntiles; i += 2) {  // ntiles even -> buffer index is static
    step(i, 0);
    step(i + 1, 1);
  }

  WAIT_ASYNCCNT(0);  // flush final async stores before S_ENDPGM
}

extern "C" void kernel_launch(void* const* d_in, const int* in_sizes, int n_in,
                              void* d_out, int out_size, void* d_ws, size_t ws_size,
                              hipStream_t stream) {
  (void)n_in; (void)out_size; (void)d_ws; (void)ws_size;
  const float* inp = (const float*)d_in[0];
  float* out = (float*)d_out;
  const int T = 4000;             // fixed by reference setup_inputs()
  const int N = in_sizes[0] / T;  // 8192 (multiple of BLOCK)
  lif_scan_kernel<<<N / BLOCK, BLOCK, 0, stream>>>(inp, out, T);
}